// MutualNearestNeighbor_83665962926605
// MI455X (gfx1250) — compile-verified
//
#include <hip/hip_runtime.h>

#define B1N 8192
#define B2N 12288
#define DN  128
#define TS  128
#define LDSP 144   // Btp row stride in float2 units (288 dwords ≡ 32 mod 64 banks -> conflict-free frag reads)

typedef __attribute__((ext_vector_type(2))) float v2f;
typedef __attribute__((ext_vector_type(8))) float v8f;

static __device__ __forceinline__ unsigned long long pk(float d, unsigned idx) {
    return ((unsigned long long)__float_as_uint(d) << 32) | (unsigned long long)idx;
}

// ---------------------------------------------------------------------------
// Kernel 1: squared norms + init packed (dist<<32|idx) min slots to ~0
// ---------------------------------------------------------------------------
__global__ __launch_bounds__(256) void mnn_prep(
    const float* __restrict__ d1, const float* __restrict__ d2,
    float* __restrict__ sq1, float* __restrict__ sq2,
    unsigned long long* __restrict__ rowp, unsigned long long* __restrict__ colp)
{
    const int tid  = threadIdx.x;
    const int lane = tid & 31;
    const int w    = tid >> 5;
    const int g    = blockIdx.x * 8 + w;   // one wave per row

    if (g < B1N) {
        float4 x = *(const float4*)(d1 + (size_t)g * DN + lane * 4);
        float s = x.x * x.x + x.y * x.y + x.z * x.z + x.w * x.w;
        for (int off = 16; off > 0; off >>= 1) s += __shfl_xor(s, off, 32);
        if (lane == 0) sq1[g] = s;
    }
    if (g < B2N) {
        float4 x = *(const float4*)(d2 + (size_t)g * DN + lane * 4);
        float s = x.x * x.x + x.y * x.y + x.z * x.z + x.w * x.w;
        for (int off = 16; off > 0; off >>= 1) s += __shfl_xor(s, off, 32);
        if (lane == 0) sq2[g] = s;
    }

    const int t = blockIdx.x * 256 + tid;
    if (t < B1N)            rowp[t]        = ~0ull;
    else if (t < B1N + B2N) colp[t - B1N]  = ~0ull;
}

// ---------------------------------------------------------------------------
// Kernel 2: fused f32-WMMA GEMM tile + row/col packed-min reduction
// Each workgroup (8 waves) owns a 128(M) x 128(N) tile; wave w owns rows
// [M0+16w, M0+16w+16). K = 128 entirely, 32 steps of 16x16x4 f32 WMMA.
// B^T tile staged in LDS as pre-paired float2 rows so each B fragment is a
// single aligned ds_load_b64 straight into the WMMA operand register pair.
// ---------------------------------------------------------------------------
__global__ __launch_bounds__(256) void mnn_gemm(
    const float* __restrict__ d1, const float* __restrict__ d2,
    const float* __restrict__ sq1, const float* __restrict__ sq2,
    unsigned long long* __restrict__ rowp, unsigned long long* __restrict__ colp)
{
    __shared__ v2f Btp[64 * LDSP];    // Btp[k/2][n] = {B[k][n], B[k+1][n]}, 72 KB

    const int tid  = threadIdx.x;
    const int lane = tid & 31;
    const int w    = tid >> 5;
    const int kh   = lane >> 4;       // K-half select (lanes 16-31 hold K+2,K+3)
    const int l15  = lane & 15;
    const int N0   = blockIdx.x * TS;
    const int M0   = blockIdx.y * TS;

    // Stage transposed B tile: each thread = one coalesced float2 global read
    // + one b64 LDS store.  idx -> (n = idx>>6, kpair = idx&63).
    #pragma unroll 4
    for (int it = 0; it < 32; ++it) {
        int idx = it * 256 + tid;                 // 0..8191 float2 elements
        int n   = idx >> 6;                       // 0..127 (tile column)
        int kp  = idx & 63;                       // float2 row (k = 2*kp)
        Btp[kp * LDSP + n] = *(const v2f*)(d2 + (size_t)(N0 + n) * DN + 2 * kp);
    }
    __syncthreads();

    v8f c[8];
    #pragma unroll
    for (int nb = 0; nb < 8; ++nb) c[nb] = (v8f){};

    // A fragment: lane L supplies row M0+16w+(L&15), K pair (k0 + 2*kh).
    const float* arow = d1 + (size_t)(M0 + 16 * w + l15) * DN;

    for (int k0 = 0; k0 < DN; k0 += 4) {
        v2f a = *(const v2f*)(arow + k0 + 2 * kh);          // global_load_b64
        const v2f* brow = &Btp[((k0 >> 1) + kh) * LDSP + l15];
        #pragma unroll
        for (int nb = 0; nb < 8; ++nb) {
            v2f b = brow[nb * 16];                           // ds_load_b64
            c[nb] = __builtin_amdgcn_wmma_f32_16x16x4_f32(
                false, a, false, b, (short)0, c[nb], false, false);
        }
    }

    // c[nb][v]: element (m = M0+16w+v+8*kh, n = N0+16*nb+l15) of A.B^T.
    float s1a[8];
    #pragma unroll
    for (int v = 0; v < 8; ++v) s1a[v] = sq1[M0 + 16 * w + v + 8 * kh];

    #pragma unroll
    for (int nb = 0; nb < 8; ++nb) {
        float s2 = sq2[N0 + nb * 16 + l15];
        #pragma unroll
        for (int v = 0; v < 8; ++v)
            c[nb][v] = fmaxf(s1a[v] + s2 - 2.0f * c[nb][v], 0.0f);  // d^2 >= 0
    }

    // Row-min over this tile's 128 columns (per row m).
    #pragma unroll
    for (int v = 0; v < 8; ++v) {
        const int m = M0 + 16 * w + v + 8 * kh;
        unsigned long long best = ~0ull;
        #pragma unroll
        for (int nb = 0; nb < 8; ++nb) {
            unsigned long long p = pk(c[nb][v], (unsigned)(N0 + nb * 16 + l15));
            best = p < best ? p : best;
        }
        #pragma unroll
        for (int off = 1; off < 16; off <<= 1) {
            unsigned long long o = __shfl_xor(best, off, 32);
            best = o < best ? o : best;
        }
        if (l15 == 0) atomicMin(&rowp[m], best);
    }

    // Col-min over this tile's 128 rows (per column n).
    #pragma unroll
    for (int nb = 0; nb < 8; ++nb) {
        const int n = N0 + nb * 16 + l15;
        unsigned long long best = ~0ull;
        #pragma unroll
        for (int v = 0; v < 8; ++v) {
            unsigned long long p = pk(c[nb][v], (unsigned)(M0 + 16 * w + v + 8 * kh));
            best = p < best ? p : best;
        }
        unsigned long long o = __shfl_xor(best, 16, 32);
        best = o < best ? o : best;
        if (kh == 0) atomicMin(&colp[n], best);
    }
}

// ---------------------------------------------------------------------------
// Kernel 3: mutual-NN check + outputs (dists | idx pairs | mutual flags)
// ---------------------------------------------------------------------------
__global__ __launch_bounds__(256) void mnn_final(
    const unsigned long long* __restrict__ rowp,
    const unsigned long long* __restrict__ colp,
    float* __restrict__ out)
{
    const int i = blockIdx.x * 256 + threadIdx.x;
    if (i >= B1N) return;
    unsigned long long rp = rowp[i];
    unsigned j  = (unsigned)(rp & 0xffffffffu);
    float d2    = __uint_as_float((unsigned)(rp >> 32));
    unsigned i2 = (unsigned)(colp[j] & 0xffffffffu);
    bool mutual = (i2 == (unsigned)i);

    out[i]               = mutual ? sqrtf(d2)  : __builtin_inff();
    out[B1N + 2 * i]     = mutual ? (float)i   : -1.0f;
    out[B1N + 2 * i + 1] = mutual ? (float)j   : -1.0f;
    out[3 * B1N + i]     = mutual ? 1.0f       : 0.0f;
}

extern "C" void kernel_launch(void* const* d_in, const int* in_sizes, int n_in,
                              void* d_out, int out_size, void* d_ws, size_t ws_size,
                              hipStream_t stream) {
    const float* d1 = (const float*)d_in[0];   // desc1: 8192 x 128 f32
    const float* d2 = (const float*)d_in[1];   // desc2: 12288 x 128 f32
    float* out = (float*)d_out;

    char* ws = (char*)d_ws;
    float* sq1 = (float*)ws;
    float* sq2 = (float*)(ws + (size_t)B1N * 4);
    unsigned long long* rowp = (unsigned long long*)(ws + (size_t)(B1N + B2N) * 4);
    unsigned long long* colp = rowp + B1N;

    mnn_prep<<<B2N / 8, 256, 0, stream>>>(d1, d2, sq1, sq2, rowp, colp);
    dim3 grid(B2N / TS, B1N / TS);             // 96 x 64 tiles
    mnn_gemm<<<grid, 256, 0, stream>>>(d1, d2, sq1, sq2, rowp, colp);
    mnn_final<<<(B1N + 255) / 256, 256, 0, stream>>>(rowp, colp, out);

    (void)in_sizes; (void)n_in; (void)out_size; (void)ws_size;
}